// MATE_28656021799674
// MI455X (gfx1250) — compile-verified
//
#include <hip/hip_runtime.h>
#include <math.h>

typedef __attribute__((ext_vector_type(2))) float v2f;
typedef __attribute__((ext_vector_type(8))) float v8f;

#define DDIM 128
#define SU_STRIDE 132   // padded row stride (floats); 132%4==0 keeps 16B alignment
#define SP_STRIDE 132

// ds_swizzle_b32 xor-butterfly: offset = {xor[14:10], or[9:5]=0, and[4:0]=0x1f}
#define ADDSWZ(v, pat) ((v) + __int_as_float(__builtin_amdgcn_ds_swizzle(__float_as_int(v), (pat))))

__device__ __forceinline__ float waveSum(float v) {
    v = ADDSWZ(v, 0x041f);   // xor 1
    v = ADDSWZ(v, 0x081f);   // xor 2
    v = ADDSWZ(v, 0x101f);   // xor 4
    v = ADDSWZ(v, 0x201f);   // xor 8
    v = ADDSWZ(v, 0x401f);   // xor 16
    return v;
}

// Single-query attention branch, online softmax, keys processed in chunks of 8
// for load/reduction/exp ILP. Lane owns 4 contiguous columns of D=128.
// Key list = S gated embedding rows followed by q itself.
template <int S>
__device__ __forceinline__ float4 attn_branch(const float4 cur, const float4 u,
                                              const float4* __restrict__ emb,
                                              const int lane) {
    constexpr int T = S + 1;                      // +1: q is the last key
    const float scale = 0.08838834764831845f;     // 1/sqrt(128)
    float4 q;
    q.x = cur.x * u.x; q.y = cur.y * u.y; q.z = cur.z * u.z; q.w = cur.w * u.w;

    float m = -INFINITY, l = 0.0f;
    float4 o; o.x = 0.f; o.y = 0.f; o.z = 0.f; o.w = 0.f;

#pragma unroll
    for (int base = 0; base < T; base += 8) {
        constexpr int CH = 8;
        const int n = (T - base < CH) ? (T - base) : CH;
        float4 k[CH];
        float  p[CH];
        // batched loads + gated keys + partial dots (independent)
#pragma unroll
        for (int j = 0; j < CH; ++j) {
            if (j < n) {
                float4 kj;
                if (base + j < S) {
                    float4 e = emb[(base + j) * 32 + lane];
                    kj.x = e.x * u.x; kj.y = e.y * u.y; kj.z = e.z * u.z; kj.w = e.w * u.w;
                } else {
                    kj = q;
                }
                k[j] = kj;
                p[j] = q.x * kj.x + q.y * kj.y + q.z * kj.z + q.w * kj.w;
            }
        }
        // 8 independent butterfly reductions
#pragma unroll
        for (int j = 0; j < CH; ++j)
            if (j < n) p[j] = waveSum(p[j]) * scale;
        // chunk max, single rescale of running state
        float cmx = p[0];
#pragma unroll
        for (int j = 1; j < CH; ++j)
            if (j < n) cmx = fmaxf(cmx, p[j]);
        float mn   = fmaxf(m, cmx);
        float corr = __expf(m - mn);               // first chunk: exp(-inf)=0
        l *= corr;
        o.x *= corr; o.y *= corr; o.z *= corr; o.w *= corr;
        // 8 independent exps, then accumulate
#pragma unroll
        for (int j = 0; j < CH; ++j) {
            if (j < n) {
                float w = __expf(p[j] - mn);
                l += w;
                o.x += w * k[j].x; o.y += w * k[j].y;
                o.z += w * k[j].z; o.w += w * k[j].w;
            }
        }
        m = mn;
    }
    float inv = 1.0f / l;
    o.x *= inv; o.y *= inv; o.z *= inv; o.w *= inv;
    return o;
}

__global__ __launch_bounds__(128)
void MATE_fused_timeattn(const float* __restrict__ user,
                         const float* __restrict__ cur_month,
                         const float* __restrict__ cur_day,
                         const float* __restrict__ cur_date,
                         const float* __restrict__ cur_hour,
                         const float* __restrict__ month_emb,
                         const float* __restrict__ day_emb,
                         const float* __restrict__ date_emb,
                         const float* __restrict__ hour_emb,
                         const float* __restrict__ W,
                         const float* __restrict__ Wmonth,
                         const float* __restrict__ Wday,
                         const float* __restrict__ Wdate,
                         const float* __restrict__ Whour,
                         float* __restrict__ out) {
    __shared__ float sU[16 * SU_STRIDE];        // user tile, 16x128 (padded)
    __shared__ float sP[5 * 16 * SP_STRIDE];    // 5 projection tiles, each 16x128 (padded)

    const int tid  = threadIdx.x;
    const int wave = tid >> 5;
    const int lane = tid & 31;
    const int b0   = blockIdx.x << 4;           // 16 batch rows per block

    // ---- stage user tile (coalesced float4 loads) ----
    const float4* u4 = (const float4*)user + (size_t)b0 * 32;
#pragma unroll
    for (int j = 0; j < 4; ++j) {
        int v = tid + 128 * j;                  // float4 index within 16x32 tile
        int row = v >> 5, col4 = v & 31;
        *(float4*)&sU[row * SU_STRIDE + col4 * 4] = u4[v];
    }
    __syncthreads();

    // ---- Phase 1: WMMA f32 16x16x4 projections: sP[mat] = userTile @ Wmat ----
    const int half = lane >> 4;                 // 0: lanes 0-15, 1: lanes 16-31
    const int l16  = lane & 15;

    for (int t = wave; t < 40; t += 4) {        // 5 matrices x 8 N-tiles
        const int mat = t >> 3;
        const int n0  = (t & 7) << 4;
        const float* Wp = (mat == 0) ? Wmonth
                        : (mat == 1) ? Wday
                        : (mat == 2) ? Wdate
                        : (mat == 3) ? Whour
                                     : W;
        v8f c = {};
#pragma unroll 4
        for (int kk = 0; kk < 32; ++kk) {
            const int k = 4 * kk + 2 * half;
            // A 16x4 f32 frag: lane l16 = row M, regs hold K=k,k+1 (b64 LDS load)
            v2f a = *(const v2f*)&sU[l16 * SU_STRIDE + k];
            // B 4x16 f32 frag: lanes index N; regs hold rows K=k,k+1
            v2f bfrag;
            bfrag.x = Wp[(size_t)k * DDIM + n0 + l16];
            bfrag.y = Wp[(size_t)(k + 1) * DDIM + n0 + l16];
            c = __builtin_amdgcn_wmma_f32_16x16x4_f32(
                    false, a, false, bfrag, (short)0, c, false, false);
        }
        // C layout: VGPR r -> (lanes 0-15: M=r, lanes 16-31: M=8+r), N = n0 + l16
        const int rowbase = half << 3;
#pragma unroll
        for (int r = 0; r < 8; ++r)
            sP[(mat * 16 + rowbase + r) * SP_STRIDE + n0 + l16] = c[r];
    }
    __syncthreads();

    // ---- Phase 2: per-row streaming attention (one wave per batch row, 4 rows/wave) ----
    const float4* cm4 = (const float4*)cur_month;
    const float4* cd4 = (const float4*)cur_day;
    const float4* ct4 = (const float4*)cur_date;
    const float4* ch4 = (const float4*)cur_hour;
    const float4* me4 = (const float4*)month_emb;
    const float4* de4 = (const float4*)day_emb;
    const float4* te4 = (const float4*)date_emb;
    const float4* he4 = (const float4*)hour_emb;
    float4* out4 = (float4*)out;

    for (int i = 0; i < 4; ++i) {
        const int lu = wave * 4 + i;
        const size_t b = (size_t)b0 + lu;

        float4 uM = *(const float4*)&sP[(0 * 16 + lu) * SP_STRIDE + 4 * lane];
        float4 uD = *(const float4*)&sP[(1 * 16 + lu) * SP_STRIDE + 4 * lane];
        float4 uT = *(const float4*)&sP[(2 * 16 + lu) * SP_STRIDE + 4 * lane];
        float4 uH = *(const float4*)&sP[(3 * 16 + lu) * SP_STRIDE + 4 * lane];
        float4 qW = *(const float4*)&sP[(4 * 16 + lu) * SP_STRIDE + 4 * lane];

        float4 km = attn_branch<11>(cm4[b * 32 + lane], uM, me4 + b * 11 * 32, lane);
        float4 kd = attn_branch< 6>(cd4[b * 32 + lane], uD, de4 + b *  6 * 32, lane);
        float4 kt = attn_branch<30>(ct4[b * 32 + lane], uT, te4 + b * 30 * 32, lane);
        float4 kh = attn_branch<23>(ch4[b * 32 + lane], uH, he4 + b * 23 * 32, lane);

        // gating: a_k = sigmoid(qW . kv_k) -- 4 independent reductions
        float p0 = waveSum(qW.x * km.x + qW.y * km.y + qW.z * km.z + qW.w * km.w);
        float p1 = waveSum(qW.x * kd.x + qW.y * kd.y + qW.z * kd.z + qW.w * kd.w);
        float p2 = waveSum(qW.x * kt.x + qW.y * kt.y + qW.z * kt.z + qW.w * kt.w);
        float p3 = waveSum(qW.x * kh.x + qW.y * kh.y + qW.z * kh.z + qW.w * kh.w);
        float a0 = 1.0f / (1.0f + __expf(-p0));
        float a1 = 1.0f / (1.0f + __expf(-p1));
        float a2 = 1.0f / (1.0f + __expf(-p2));
        float a3 = 1.0f / (1.0f + __expf(-p3));

        float4 o;
        o.x = a0 * km.x + a1 * kd.x + a2 * kt.x + a3 * kh.x;
        o.y = a0 * km.y + a1 * kd.y + a2 * kt.y + a3 * kh.y;
        o.z = a0 * km.z + a1 * kd.z + a2 * kt.z + a3 * kh.z;
        o.w = a0 * km.w + a1 * kd.w + a2 * kt.w + a3 * kh.w;
        out4[b * 32 + lane] = o;
    }
}

extern "C" void kernel_launch(void* const* d_in, const int* in_sizes, int n_in,
                              void* d_out, int out_size, void* d_ws, size_t ws_size,
                              hipStream_t stream) {
    const float* user      = (const float*)d_in[0];
    const float* cur_month = (const float*)d_in[1];
    const float* cur_day   = (const float*)d_in[2];
    const float* cur_date  = (const float*)d_in[3];
    const float* cur_hour  = (const float*)d_in[4];
    const float* month_emb = (const float*)d_in[5];
    const float* day_emb   = (const float*)d_in[6];
    const float* date_emb  = (const float*)d_in[7];
    const float* hour_emb  = (const float*)d_in[8];
    const float* W         = (const float*)d_in[9];
    const float* Wmonth    = (const float*)d_in[10];
    const float* Wday      = (const float*)d_in[11];
    const float* Wdate     = (const float*)d_in[12];
    const float* Whour     = (const float*)d_in[13];
    float* out = (float*)d_out;

    const int Btot = in_sizes[0] / DDIM;        // 16384
    dim3 grid(Btot / 16), block(128);
    MATE_fused_timeattn<<<grid, block, 0, stream>>>(
        user, cur_month, cur_day, cur_date, cur_hour,
        month_emb, day_emb, date_emb, hour_emb,
        W, Wmonth, Wday, Wdate, Whour, out);
}